// TemporalBlock_70119636074539
// MI455X (gfx1250) — compile-verified
//
#include <hip/hip_runtime.h>

namespace {

constexpr int  Bq  = 2;
constexpr int  T   = 8;
constexpr int  Tt  = 9;                 // T_
constexpr int  Hh  = 24;
constexpr int  Ww  = 24;
constexpr int  HW  = 576;
constexpr int  Dd  = 512;
constexpr int  NH  = 8;
constexpr int  HD  = 64;
constexpr int  MS  = Bq * T * HW;       // 9216 visual tokens
constexpr int  MT  = Bq * Tt * HW;      // 10368 total tokens
constexpr size_t BIG = (size_t)MT * Dd; // 5,308,416 floats

typedef __attribute__((ext_vector_type(2))) float v2f;
typedef __attribute__((ext_vector_type(8))) float v8f;

enum { GF_BIAS = 1, GF_GELU = 2, GF_ACC = 4, GF_RESID = 8 };

// B panel in LDS, k-pair interleaved: element (k, col) of a 16x64 chunk lives at
//   blds[buf][(k>>1)*PS + col*2 + (k&1)]
// so the WMMA B fragment {B[k][col], B[k+1][col]} is one aligned 8-byte ds_load_b64.
// PS = 160 floats => pair-row stride is 32 banks, putting the two lane-halves
// (which read pair p and p+1) on complementary 32-bank sets: conflict-free.
constexpr int PS = 160;

__device__ __forceinline__ float waveRedSum(float v) {
#pragma unroll
  for (int o = 16; o > 0; o >>= 1) v += __shfl_xor(v, o, 32);
  return v;
}

// ---------------- fp32 WMMA GEMM: C = A(MxK) * B(KxN) (+bias)(+gelu)(+resid)(+=C)
// Block = 128 threads = 4 waves stacked along M (64 rows); each wave computes a
// 16(M) x 64(N) strip as 4 accumulators, reusing its A fragment across 4 WMMAs.
// The 16x64 B panel per K-chunk is staged in double-buffered LDS, shared by all
// waves, one barrier per chunk.  Requirements: M%64==0, N%64==0, K%16==0.
__global__ void gemm_wmma_f32(const float* __restrict__ A, int lda,
                              const float* __restrict__ Bm, int ldb,
                              const float* __restrict__ bias,
                              const float* __restrict__ resid,
                              float* __restrict__ C, int ldc,
                              int K, int flags) {
  __shared__ float blds[2][8 * PS];

  const int lane = threadIdx.x & 31;
  const int wave = threadIdx.x >> 5;
  const int l15  = lane & 15;
  const int lh   = lane >> 4;
  const int mBase = blockIdx.y * 64 + wave * 16;
  const int nBase = blockIdx.x * 64;

  const float* Ar = A + (size_t)(mBase + l15) * lda;   // wave's A rows (m = l15-local)

  // B staging pattern: 128 threads, each copies rows r0 and r0+8, cols c4..c4+3
  const int r0 = threadIdx.x >> 4;          // 0..7
  const int c4 = (threadIdx.x & 15) * 4;    // 0..60

  v8f c0 = {}, c1 = {}, c2 = {}, c3 = {};

  int buf = 0;
  for (int k0 = 0; k0 < K; k0 += 16, buf ^= 1) {
    float* bl = blds[buf];
#pragma unroll
    for (int pass = 0; pass < 2; ++pass) {
      const int r = r0 + 8 * pass;
      const float4 f = *(const float4*)(Bm + (size_t)(k0 + r) * ldb + nBase + c4);
      float* dst = bl + (r >> 1) * PS + (r & 1) + c4 * 2;
      dst[0] = f.x; dst[2] = f.y; dst[4] = f.z; dst[6] = f.w;
    }
    __syncthreads();                        // store->compute; also orders next reuse

    if (k0 + 16 < K) {                      // prefetch next A chunk (global_prefetch_b8)
      __builtin_prefetch(Ar + k0 + 16, 0, 1);
    }

#pragma unroll
    for (int kk = 0; kk < 4; ++kk) {
      const int kl = kk * 4 + lh * 2;       // even local K row for this lane-half
      const int ka = k0 + kl;
      v2f a = { Ar[ka], Ar[ka + 1] };       // one b64 load, reused by 4 WMMAs
      const float* bp = bl + (kl >> 1) * PS;
      {
        v2f b = *(const v2f*)(bp + l15 * 2);
        c0 = __builtin_amdgcn_wmma_f32_16x16x4_f32(false, a, false, b, (short)0, c0, false, false);
      }
      {
        v2f b = *(const v2f*)(bp + (16 + l15) * 2);
        c1 = __builtin_amdgcn_wmma_f32_16x16x4_f32(false, a, false, b, (short)0, c1, false, false);
      }
      {
        v2f b = *(const v2f*)(bp + (32 + l15) * 2);
        c2 = __builtin_amdgcn_wmma_f32_16x16x4_f32(false, a, false, b, (short)0, c2, false, false);
      }
      {
        v2f b = *(const v2f*)(bp + (48 + l15) * 2);
        c3 = __builtin_amdgcn_wmma_f32_16x16x4_f32(false, a, false, b, (short)0, c3, false, false);
      }
    }
  }

  // Epilogue: C layout per tile: m = j + 8*lh, n = 16*t + l15
#pragma unroll
  for (int t = 0; t < 4; ++t) {
    const v8f c = (t == 0) ? c0 : (t == 1) ? c1 : (t == 2) ? c2 : c3;
    const int n = nBase + 16 * t + l15;
    const float bval = (flags & GF_BIAS) ? bias[n] : 0.0f;
#pragma unroll
    for (int j = 0; j < 8; ++j) {
      const int m = mBase + j + lh * 8;
      float v = c[j] + bval;
      if (flags & GF_GELU) v = 0.5f * v * (1.0f + erff(v * 0.70710678118654752f));
      const size_t idx = (size_t)m * ldc + n;
      if (flags & GF_RESID) v += resid[idx];
      if (flags & GF_ACC)   v += C[idx];
      C[idx] = v;
    }
  }
}

// ---------------- layernorm: one wave per row of 512.  mapMode 1 = visual rows of query.
__global__ void ln_kernel(const float* __restrict__ X,
                          const float* __restrict__ gam,
                          const float* __restrict__ bet,
                          float* __restrict__ Y, int M, int mapMode) {
  const int lane = threadIdx.x & 31;
  const int row = blockIdx.x * 8 + (threadIdx.x >> 5);
  if (row >= M) return;
  int src = row;
  if (mapMode == 1) src = row + (row / (T * HW) + 1) * HW;  // skip CLS frame per batch
  const float* xr = X + (size_t)src * Dd;
  float v[16], s = 0.f, sq = 0.f;
#pragma unroll
  for (int k = 0; k < 4; ++k) {
    const float4 f = *(const float4*)(xr + 4 * (lane + 32 * k));
    v[4*k+0]=f.x; v[4*k+1]=f.y; v[4*k+2]=f.z; v[4*k+3]=f.w;
    s  += f.x + f.y + f.z + f.w;
    sq += f.x*f.x + f.y*f.y + f.z*f.z + f.w*f.w;
  }
  s = waveRedSum(s); sq = waveRedSum(sq);
  const float mean = s * (1.0f / Dd);
  float var = sq * (1.0f / Dd) - mean * mean;
  var = var < 0.f ? 0.f : var;
  const float rstd = rsqrtf(var + 1e-5f);
  float* yr = Y + (size_t)row * Dd;
#pragma unroll
  for (int k = 0; k < 4; ++k) {
    const int d0 = 4 * (lane + 32 * k);
    float4 o;
    o.x = (v[4*k+0]-mean)*rstd*gam[d0+0]+bet[d0+0];
    o.y = (v[4*k+1]-mean)*rstd*gam[d0+1]+bet[d0+1];
    o.z = (v[4*k+2]-mean)*rstd*gam[d0+2]+bet[d0+2];
    o.w = (v[4*k+3]-mean)*rstd*gam[d0+3]+bet[d0+3];
    *(float4*)(yr + d0) = o;
  }
}

// ---------------- mean over HW of frame 0 (query CLS frame or x2 frame 0)
__global__ void mean0_kernel(const float* __restrict__ X, float* __restrict__ out) {
  const int b = blockIdx.x, d = threadIdx.x;  // 256 threads: d and d+256
  const float* base = X + (size_t)b * Tt * HW * Dd;
  float s0 = 0.f, s1 = 0.f;
  for (int hw = 0; hw < HW; ++hw) {
    s0 += base[(size_t)hw * Dd + d];
    s1 += base[(size_t)hw * Dd + d + 256];
  }
  out[b * Dd + d]       = s0 * (1.0f / HW);
  out[b * Dd + d + 256] = s1 * (1.0f / HW);
}

__global__ void bcast0_kernel(float* __restrict__ X, const float* __restrict__ avg) {
  const size_t i = (size_t)blockIdx.x * blockDim.x + threadIdx.x;  // Bq*HW*Dd
  if (i >= (size_t)Bq * HW * Dd) return;
  const int d = i % Dd;
  const size_t r = i / Dd;
  const int b = r / HW, hw = r % HW;
  X[((size_t)b * Tt * HW + hw) * Dd + d] = avg[b * Dd + d];
}

// ---------------- tiny dense projection: out[r][n] = bias[n] + sum_d in[r][d]*W[d][n]
__global__ void small_proj(const float* __restrict__ in,
                           const float* __restrict__ Wt,
                           const float* __restrict__ bias,
                           float* __restrict__ out, int R) {
  const int idx = blockIdx.x * blockDim.x + threadIdx.x;
  if (idx >= R * Dd) return;
  const int r = idx / Dd, n = idx % Dd;
  const float* ir = in + (size_t)r * Dd;
  float s = bias[n];
  for (int d = 0; d < Dd; ++d) s += ir[d] * Wt[(size_t)d * Dd + n];
  out[idx] = s;
}

// ---------------- spatial attention: one wave per (token, head)
__global__ void spatial_attn(const float* __restrict__ Q,  const float* __restrict__ Kk,
                             const float* __restrict__ Vv,
                             const float* __restrict__ clsK, const float* __restrict__ clsV,
                             const float* __restrict__ padK, const float* __restrict__ padV,
                             float* __restrict__ Out) {
  const int lane = threadIdx.x & 31;
  const int task = blockIdx.x * 8 + (threadIdx.x >> 5);
  if (task >= MS * NH) return;
  const int h = task & 7;
  const int n = task >> 3;
  const int bcls = n % Bq;                 // torch .repeat tiling quirk
  const int hw = n % HW;
  const int y = hw / Ww, x = hw % Ww;
  const int frameBase = n - hw;

  const float* qp = Q + (size_t)n * Dd + h * HD;
  const float q0 = qp[lane], q1 = qp[lane + 32];

  float s[10];
  const float* vptr[10];

  { // kv[0] = CLS; masked by window-pos-0 padding (mask concat quirk)
    const float* kp = clsK + bcls * Dd + h * HD;
    float d = waveRedSum(q0 * kp[lane] + q1 * kp[lane + 32]) * 0.125f;
    const bool m0 = (y == 0) || (x == 0);
    s[0] = m0 ? -1e9f : d;
    vptr[0] = clsV + bcls * Dd + h * HD;
  }
#pragma unroll
  for (int p = 0; p < 9; ++p) {            // kv[p+1] = neighbor window-pos p
    const int ny = y + p / 3 - 1, nx = x + p % 3 - 1;
    const bool valid = (ny >= 0) && (ny < Hh) && (nx >= 0) && (nx < Ww);
    const size_t roff = (size_t)(frameBase + ny * Ww + nx) * Dd + h * HD;
    const float* kp = valid ? (Kk + roff) : (padK + h * HD);
    const float* vp = valid ? (Vv + roff) : (padV + h * HD);
    float d = waveRedSum(q0 * kp[lane] + q1 * kp[lane + 32]) * 0.125f;
    bool masked = false;                   // masked by window-pos (p+1); kv[9] never masked
    if (p < 8) {
      const int my = y + (p + 1) / 3 - 1, mx = x + (p + 1) % 3 - 1;
      masked = !((my >= 0) && (my < Hh) && (mx >= 0) && (mx < Ww));
    }
    s[p + 1] = masked ? -1e9f : d;
    vptr[p + 1] = vp;
  }

  float mx = s[0];
#pragma unroll
  for (int i = 1; i < 10; ++i) mx = fmaxf(mx, s[i]);
  float pb[10], sum = 0.f;
#pragma unroll
  for (int i = 0; i < 10; ++i) { pb[i] = expf(s[i] - mx); sum += pb[i]; }
  const float inv = 1.0f / sum;
  float o0 = 0.f, o1 = 0.f;
#pragma unroll
  for (int i = 0; i < 10; ++i) {
    const float w = pb[i] * inv;
    o0 += w * vptr[i][lane];
    o1 += w * vptr[i][lane + 32];
  }
  float* op = Out + (size_t)n * Dd + h * HD;
  op[lane] = o0; op[lane + 32] = o1;
}

// ---------------- temporal attention: one wave per (b,hw,head,tq); also emits logits
__global__ void temporal_attn(const float* __restrict__ Q, const float* __restrict__ Kk,
                              const float* __restrict__ Vv,
                              const unsigned char* __restrict__ tmask,
                              float* __restrict__ Out, float* __restrict__ Logits) {
  const int lane = threadIdx.x & 31;
  const int task = blockIdx.x * 8 + (threadIdx.x >> 5);
  if (task >= Bq * HW * NH * Tt) return;
  const int tq = task % Tt;
  const int t2 = task / Tt;
  const int h = t2 & 7;
  const int g = t2 >> 3;                 // b*HW + hw
  const int b = g / HW, hw = g % HW;

  const size_t rowq = (size_t)(b * Tt + tq) * HW + hw;
  const float* qp = Q + rowq * Dd + h * HD;
  const float q0 = qp[lane], q1 = qp[lane + 32];
  const size_t lbase = ((size_t)g * NH + h) * (Tt * Tt) + (size_t)tq * Tt;

  float s[Tt];
#pragma unroll
  for (int tk = 0; tk < Tt; ++tk) {
    const size_t rk = (size_t)(b * Tt + tk) * HW + hw;
    const float* kp = Kk + rk * Dd + h * HD;
    float d = waveRedSum(q0 * kp[lane] + q1 * kp[lane + 32]) * 0.125f;
    if (tmask[tq * Tt + tk]) d = -1e9f;
    s[tk] = d;
    if (lane == tk) Logits[lbase + tk] = d;  // post-mask, pre-softmax
  }
  float mx = s[0];
#pragma unroll
  for (int i = 1; i < Tt; ++i) mx = fmaxf(mx, s[i]);
  float pb[Tt], sum = 0.f;
#pragma unroll
  for (int i = 0; i < Tt; ++i) { pb[i] = expf(s[i] - mx); sum += pb[i]; }
  const float inv = 1.0f / sum;
  float o0 = 0.f, o1 = 0.f;
#pragma unroll
  for (int tk = 0; tk < Tt; ++tk) {
    const float w = pb[tk] * inv;
    const float* vp = Vv + ((size_t)(b * Tt + tk) * HW + hw) * Dd + h * HD;
    o0 += w * vp[lane];
    o1 += w * vp[lane + 32];
  }
  float* op = Out + rowq * Dd + h * HD;
  op[lane] = o0; op[lane + 32] = o1;
}

// ---------------- x = concat(query CLS frame, sx + visual)
__global__ void scatter_x(const float* __restrict__ query, const float* __restrict__ sx,
                          float* __restrict__ xout) {
  const size_t i = (size_t)blockIdx.x * blockDim.x + threadIdx.x;
  if (i >= BIG) return;
  const size_t row = i / Dd;
  const int d = i % Dd;
  const int bt = row / HW;
  const int tt = bt % Tt, b = bt / Tt;
  float v = query[i];
  if (tt > 0) {
    const size_t srow = (size_t)(b * T + (tt - 1)) * HW + (row % HW);
    v += sx[srow * Dd + d];
  }
  xout[i] = v;
}

__global__ void initout_kernel(const float* __restrict__ x2, const float* __restrict__ b2,
                               float* __restrict__ out) {
  const size_t i = (size_t)blockIdx.x * blockDim.x + threadIdx.x;
  if (i >= BIG) return;
  out[i] = x2[i] + b2[i % Dd];
}

} // namespace

extern "C" void kernel_launch(void* const* d_in, const int* in_sizes, int n_in,
                              void* d_out, int out_size, void* d_ws, size_t ws_size,
                              hipStream_t stream) {
  const float* query = (const float*)d_in[0];
  const unsigned char* tmask = (const unsigned char*)d_in[2];
  const float* s_wq = (const float*)d_in[3];  const float* s_bq = (const float*)d_in[4];
  const float* s_wk = (const float*)d_in[5];  const float* s_bk = (const float*)d_in[6];
  const float* s_wv = (const float*)d_in[7];  const float* s_bv = (const float*)d_in[8];
  const float* s_wo = (const float*)d_in[9];  const float* s_bo = (const float*)d_in[10];
  const float* t_wq = (const float*)d_in[11]; const float* t_bq = (const float*)d_in[12];
  const float* t_wk = (const float*)d_in[13]; const float* t_bk = (const float*)d_in[14];
  const float* t_wv = (const float*)d_in[15]; const float* t_bv = (const float*)d_in[16];
  const float* t_wo = (const float*)d_in[17]; const float* t_bo = (const float*)d_in[18];
  const float* ln_s_g = (const float*)d_in[19]; const float* ln_s_b = (const float*)d_in[20];
  const float* ln_t_g = (const float*)d_in[21]; const float* ln_t_b = (const float*)d_in[22];
  const float* ln_m_g = (const float*)d_in[23]; const float* ln_m_b = (const float*)d_in[24];
  const float* mlp_w1 = (const float*)d_in[25]; const float* mlp_b1 = (const float*)d_in[26];
  const float* mlp_w2 = (const float*)d_in[27]; const float* mlp_b2 = (const float*)d_in[28];

  float* out_x = (float*)d_out;                 // (B,T_,HW,D)
  float* out_logits = out_x + BIG;              // (B,HW,NH,T_,T_)

  float* ws = (float*)d_ws;
  float* cls_mean = ws;          // 1024
  float* cls_norm = ws + 1024;   // 1024
  float* clsK = ws + 2048;       // 1024
  float* clsV = ws + 3072;       // 1024
  float* padK = ws + 4096;       // 512
  float* padV = ws + 4608;       // 512
  float* avg  = ws + 5120;       // 1024
  float* bufA = ws + 8192;
  float* bufB = bufA + BIG;
  float* bufC = bufB + BIG;
  float* bufD = bufC + BIG;
  float* bufE = out_x;           // reuse output-x region as temporal V until MLP init

  const dim3 gblk(128);
  auto gemm = [&](const float* A, int lda, const float* Bm, int ldb,
                  const float* bias, const float* resid,
                  float* C, int ldc, int M, int N, int K, int flags) {
    dim3 grid(N / 64, M / 64);
    gemm_wmma_f32<<<grid, gblk, 0, stream>>>(A, lda, Bm, ldb, bias, resid, C, ldc, K, flags);
  };

  // ---- spatial branch ----
  mean0_kernel<<<dim3(Bq), dim3(256), 0, stream>>>(query, cls_mean);
  ln_kernel<<<dim3(1), dim3(256), 0, stream>>>(cls_mean, ln_s_g, ln_s_b, cls_norm, Bq, 0);
  small_proj<<<dim3((Bq * Dd + 255) / 256), dim3(256), 0, stream>>>(cls_norm, s_wk, s_bk, clsK, Bq);
  small_proj<<<dim3((Bq * Dd + 255) / 256), dim3(256), 0, stream>>>(cls_norm, s_wv, s_bv, clsV, Bq);
  // padded-token kv: layernorm(0) == ln_s_b, projected
  small_proj<<<dim3((Dd + 255) / 256), dim3(256), 0, stream>>>(ln_s_b, s_wk, s_bk, padK, 1);
  small_proj<<<dim3((Dd + 255) / 256), dim3(256), 0, stream>>>(ln_s_b, s_wv, s_bv, padV, 1);

  ln_kernel<<<dim3(MS / 8), dim3(256), 0, stream>>>(query, ln_s_g, ln_s_b, bufA, MS, 1); // xn_s
  gemm(bufA, Dd, s_wq, Dd, s_bq, nullptr, bufB, Dd, MS, Dd, Dd, GF_BIAS);  // Q
  gemm(bufA, Dd, s_wk, Dd, s_bk, nullptr, bufC, Dd, MS, Dd, Dd, GF_BIAS);  // K
  gemm(bufA, Dd, s_wv, Dd, s_bv, nullptr, bufD, Dd, MS, Dd, Dd, GF_BIAS);  // V
  spatial_attn<<<dim3(MS * NH / 8), dim3(256), 0, stream>>>(bufB, bufC, bufD,
                                                            clsK, clsV, padK, padV, bufA);
  gemm(bufA, Dd, s_wo, Dd, s_bo, nullptr, bufB, Dd, MS, Dd, Dd, GF_BIAS);  // sx
  scatter_x<<<dim3((unsigned)((BIG + 255) / 256)), dim3(256), 0, stream>>>(query, bufB, bufC); // x

  // ---- temporal branch ----
  ln_kernel<<<dim3(MT / 8), dim3(256), 0, stream>>>(bufC, ln_t_g, ln_t_b, bufD, MT, 0);  // xn_t
  gemm(bufD, Dd, t_wq, Dd, t_bq, nullptr, bufA, Dd, MT, Dd, Dd, GF_BIAS);  // Qt
  gemm(bufD, Dd, t_wk, Dd, t_bk, nullptr, bufB, Dd, MT, Dd, Dd, GF_BIAS);  // Kt
  gemm(bufD, Dd, t_wv, Dd, t_bv, nullptr, bufE, Dd, MT, Dd, Dd, GF_BIAS);  // Vt (in d_out region)
  temporal_attn<<<dim3(Bq * HW * NH * Tt / 8), dim3(256), 0, stream>>>(bufA, bufB, bufE,
                                                                       tmask, bufD, out_logits);
  gemm(bufD, Dd, t_wo, Dd, t_bo, bufC, bufA, Dd, MT, Dd, Dd, GF_BIAS | GF_RESID); // x2 = tx + x

  mean0_kernel<<<dim3(Bq), dim3(256), 0, stream>>>(bufA, avg);
  bcast0_kernel<<<dim3((Bq * HW * Dd + 255) / 256), dim3(256), 0, stream>>>(bufA, avg);

  // ---- MLP residual (chunked over K=2048 in 4 slabs of 512, accumulating into d_out) ----
  ln_kernel<<<dim3(MT / 8), dim3(256), 0, stream>>>(bufA, ln_m_g, ln_m_b, bufB, MT, 0);  // xn_m
  initout_kernel<<<dim3((unsigned)((BIG + 255) / 256)), dim3(256), 0, stream>>>(bufA, mlp_b2, out_x);
  for (int c = 0; c < 4; ++c) {
    gemm(bufB, Dd, mlp_w1 + c * 512, 2048, mlp_b1 + c * 512, nullptr,
         bufD, 512, MT, 512, Dd, GF_BIAS | GF_GELU);                       // h chunk
    gemm(bufD, 512, mlp_w2 + (size_t)c * 512 * 512, 512, nullptr, nullptr,
         out_x, Dd, MT, Dd, 512, GF_ACC);                                  // out += h_c @ w2_c
  }
}